// SpatialRGCN_89094801588564
// MI455X (gfx1250) — compile-verified
//
#include <hip/hip_runtime.h>
#include <hip/hip_bf16.h>

typedef __attribute__((ext_vector_type(2))) float v2f;
typedef __attribute__((ext_vector_type(8))) float v8f;

#define HID   128
#define NREL  3
#define NW    4          // 3 relation matrices + root
#define DSQ   (HID*HID)  // 16384
#define LN_EPS 1e-5f

// ---------------------------------------------------------------------------
// K1: build TRANSPOSED weights:  WT4[w][col][k] = W[w][k][col]
//     where W[w] = (w<3) ? comp[w,0]*bases[0] + comp[w,1]*bases[1] : root.
//     Transposed storage makes each WMMA B-fragment (W[k][col], W[k+1][col])
//     a contiguous 8-byte load in the GEMM hot loop.
// ---------------------------------------------------------------------------
__global__ __launch_bounds__(256) void rgcn_make_w(const float* __restrict__ bases_l,
                                                   const float* __restrict__ comp_l,
                                                   const float* __restrict__ root_l,
                                                   float* __restrict__ WT4) {
    int i   = blockIdx.x * 256 + threadIdx.x;   // 0 .. NW*DSQ-1
    int w   = i >> 14;
    int col = (i >> 7) & (HID - 1);
    int k   = i & (HID - 1);
    int ij  = k * HID + col;                    // row-major source index
    float v;
    if (w < NREL) {
        v = comp_l[w * 2 + 0] * bases_l[ij] + comp_l[w * 2 + 1] * bases_l[DSQ + ij];
    } else {
        v = root_l[ij];
    }
    WT4[i] = v;   // coalesced write of WT
}

// ---------------------------------------------------------------------------
// K2: xw4[w][n][:] = h[n][:] @ W[w]  for w = 0..3, via V_WMMA_F32_16X16X4_F32
//     block = 256 threads = 8 waves; block owns 16 rows (LDS-staged), each
//     wave owns one 16-col tile; 4 independent accumulators (one per weight
//     matrix) so consecutive WMMAs have no RAW dependency.
// ---------------------------------------------------------------------------
__global__ __launch_bounds__(256) void rgcn_gemm(const float* __restrict__ h,
                                                 const float* __restrict__ WT4,
                                                 float* __restrict__ xw4,
                                                 int N) {
    __shared__ float hs[16][HID + 4];   // pad: A-frag reads hit distinct banks
    const int row0 = blockIdx.x * 16;

    for (int idx = threadIdx.x; idx < 16 * HID; idx += 256) {
        int r = idx >> 7, c = idx & (HID - 1);
        int gr = row0 + r;
        hs[r][c] = (gr < N) ? h[(size_t)gr * HID + c] : 0.0f;
    }
    __syncthreads();

    const int wave  = threadIdx.x >> 5;
    const int lane  = threadIdx.x & 31;
    const int col0  = wave * 16;
    const int mrow  = lane & 15;     // A row / B-C column within tile
    const int khalf = lane >> 4;     // lanes 16-31 handle K+2 / rows M+8

    v8f acc0 = {}, acc1 = {}, acc2 = {}, acc3 = {};
    // WT layout: WT4[w*DSQ + col*HID + k]; this lane's column is col0+mrow.
    const float* Wc = WT4 + (size_t)(col0 + mrow) * HID;

#pragma unroll 8
    for (int k0 = 0; k0 < HID; k0 += 4) {
        const int k = k0 + 2 * khalf;
        // A 16x4 fragment: VGPR0 = K(k), VGPR1 = K(k+1)
        v2f a;
        a.x = hs[mrow][k];
        a.y = hs[mrow][k + 1];
        // B 4x16 fragments: rows k, k+1 at this lane's column -> one b64 load
        v2f b0 = *(const v2f*)(Wc + 0 * DSQ + k);
        v2f b1 = *(const v2f*)(Wc + 1 * DSQ + k);
        v2f b2 = *(const v2f*)(Wc + 2 * DSQ + k);
        v2f b3 = *(const v2f*)(Wc + 3 * DSQ + k);
        acc0 = __builtin_amdgcn_wmma_f32_16x16x4_f32(false, a, false, b0, (short)0, acc0, false, false);
        acc1 = __builtin_amdgcn_wmma_f32_16x16x4_f32(false, a, false, b1, (short)0, acc1, false, false);
        acc2 = __builtin_amdgcn_wmma_f32_16x16x4_f32(false, a, false, b2, (short)0, acc2, false, false);
        acc3 = __builtin_amdgcn_wmma_f32_16x16x4_f32(false, a, false, b3, (short)0, acc3, false, false);
    }

    // C/D layout: VGPR v -> row (v + 8*khalf), col = col0 + mrow
    const size_t nd = (size_t)N * HID;
#pragma unroll
    for (int v = 0; v < 8; ++v) {
        int gr = row0 + v + 8 * khalf;
        if (gr < N) {
            size_t base = (size_t)gr * HID + col0 + mrow;
            xw4[0 * nd + base] = acc0[v];
            xw4[1 * nd + base] = acc1[v];
            xw4[2 * nd + base] = acc2[v];
            xw4[3 * nd + base] = acc3[v];
        }
    }
}

// ---------------------------------------------------------------------------
// K3: per-edge gather + atomic scatter into (rel, dst) segments. One wave/edge.
// ---------------------------------------------------------------------------
__global__ __launch_bounds__(256) void rgcn_scatter(const int* __restrict__ src,
                                                    const int* __restrict__ dst,
                                                    const int* __restrict__ etype,
                                                    const float* __restrict__ xw4,
                                                    float* __restrict__ sums,
                                                    float* __restrict__ cnt,
                                                    int N, int E) {
    int e    = (blockIdx.x * 256 + threadIdx.x) >> 5;
    int lane = threadIdx.x & 31;
    if (e >= E) return;
    int r = etype[e], s = src[e], t = dst[e];
    const float4* row = (const float4*)(xw4 + ((size_t)r * N + s) * HID);
    float4 v = row[lane];
    float* o = sums + ((size_t)r * N + t) * HID + lane * 4;
    atomicAdd(o + 0, v.x);
    atomicAdd(o + 1, v.y);
    atomicAdd(o + 2, v.z);
    atomicAdd(o + 3, v.w);
    if (lane == 0) atomicAdd(cnt + (size_t)r * N + t, 1.0f);
}

// ---------------------------------------------------------------------------
// K4: agg = sum_r sums/max(cnt,1); + root term + bias; LayerNorm; ELU; residual
//     One wave per node, 4 channels per lane, shfl_xor reduction (wave32).
// ---------------------------------------------------------------------------
__device__ __forceinline__ float ln_elu_res(float v, float mu, float rstd,
                                            float w, float b, float hres) {
    float y = (v - mu) * rstd * w + b;
    y = (y > 0.0f) ? y : expm1f(y);
    return y + hres;
}

__global__ __launch_bounds__(256) void rgcn_finalize(const float* __restrict__ sums,
                                                     const float* __restrict__ cnt,
                                                     const float* __restrict__ xroot,
                                                     const float* __restrict__ hin,
                                                     const float* __restrict__ bias,
                                                     const float* __restrict__ lnw,
                                                     const float* __restrict__ lnb,
                                                     float* __restrict__ hout,
                                                     int N) {
    int node = (blockIdx.x * 256 + threadIdx.x) >> 5;
    int lane = threadIdx.x & 31;
    if (node >= N) return;

    const size_t nd     = (size_t)N * HID;
    const size_t rowoff = (size_t)node * HID;
    const int    col    = lane * 4;

    float4 v = make_float4(0.f, 0.f, 0.f, 0.f);
#pragma unroll
    for (int r = 0; r < NREL; ++r) {
        float inv = 1.0f / fmaxf(cnt[(size_t)r * N + node], 1.0f);
        float4 sv = ((const float4*)(sums + r * nd + rowoff))[lane];
        v.x += sv.x * inv; v.y += sv.y * inv; v.z += sv.z * inv; v.w += sv.w * inv;
    }
    float4 rv = ((const float4*)(xroot + rowoff))[lane];
    v.x += rv.x + bias[col + 0];
    v.y += rv.y + bias[col + 1];
    v.z += rv.z + bias[col + 2];
    v.w += rv.w + bias[col + 3];

    float s  = v.x + v.y + v.z + v.w;
    float sq = v.x * v.x + v.y * v.y + v.z * v.z + v.w * v.w;
#pragma unroll
    for (int m = 16; m >= 1; m >>= 1) {
        s  += __shfl_xor(s,  m, 32);
        sq += __shfl_xor(sq, m, 32);
    }
    float mu   = s * (1.0f / HID);
    float var  = sq * (1.0f / HID) - mu * mu;
    float rstd = rsqrtf(var + LN_EPS);

    float4 hv = ((const float4*)(hin + rowoff))[lane];
    float4 o;
    o.x = ln_elu_res(v.x, mu, rstd, lnw[col + 0], lnb[col + 0], hv.x);
    o.y = ln_elu_res(v.y, mu, rstd, lnw[col + 1], lnb[col + 1], hv.y);
    o.z = ln_elu_res(v.z, mu, rstd, lnw[col + 2], lnb[col + 2], hv.z);
    o.w = ln_elu_res(v.w, mu, rstd, lnw[col + 3], lnb[col + 3], hv.w);
    ((float4*)(hout + rowoff))[lane] = o;
}

// ---------------------------------------------------------------------------
// Host launcher (graph-capture safe: only kernels + hipMemsetAsync on stream)
// ---------------------------------------------------------------------------
extern "C" void kernel_launch(void* const* d_in, const int* in_sizes, int n_in,
                              void* d_out, int out_size, void* d_ws, size_t ws_size,
                              hipStream_t stream) {
    const float* x     = (const float*)d_in[0];
    const int*   ei    = (const int*)d_in[1];   // (2,E): src row then dst row
    const int*   et    = (const int*)d_in[2];   // (E,)
    const float* bases = (const float*)d_in[3]; // (L,B,d,d)
    const float* comp  = (const float*)d_in[4]; // (L,R,B)
    const float* root  = (const float*)d_in[5]; // (L,d,d)
    const float* bias  = (const float*)d_in[6]; // (L,d)
    const float* lnw   = (const float*)d_in[7]; // (L,d)
    const float* lnb   = (const float*)d_in[8]; // (L,d)

    const int N = in_sizes[0] / HID;
    const int E = in_sizes[1] / 2;
    const int* srcp = ei;
    const int* dstp = ei + E;

    const size_t nd = (size_t)N * HID;
    float* ws   = (float*)d_ws;
    float* WT4  = ws;                       // NW*DSQ (transposed weights)
    float* xw4  = WT4 + (size_t)NW * DSQ;   // NW*nd
    float* sums = xw4 + (size_t)NW * nd;    // NREL*nd
    float* cnts = sums + (size_t)NREL * nd; // NREL*N (contiguous after sums)
    float* h1   = cnts + (size_t)NREL * N;  // nd

    float* out = (float*)d_out;

    const int gemm_blocks = (N + 15) / 16;
    const int scat_blocks = (E + 7) / 8;
    const int fin_blocks  = (N + 7) / 8;

    const float* hin = x;
    for (int l = 0; l < 2; ++l) {
        float* hout = (l == 0) ? h1 : out;

        rgcn_make_w<<<(NW * DSQ) / 256, 256, 0, stream>>>(
            bases + (size_t)l * 2 * DSQ, comp + (size_t)l * NREL * 2,
            root + (size_t)l * DSQ, WT4);

        rgcn_gemm<<<gemm_blocks, 256, 0, stream>>>(hin, WT4, xw4, N);

        hipMemsetAsync(sums, 0, (NREL * nd + (size_t)NREL * N) * sizeof(float), stream);

        rgcn_scatter<<<scat_blocks, 256, 0, stream>>>(srcp, dstp, et, xw4, sums, cnts, N, E);

        rgcn_finalize<<<fin_blocks, 256, 0, stream>>>(
            sums, cnts, xw4 + 3 * nd, hin,
            bias + (size_t)l * HID, lnw + (size_t)l * HID, lnb + (size_t)l * HID,
            hout, N);

        hin = hout;
    }
}